// GraphSage_49752901156948
// MI455X (gfx1250) — compile-verified
//
#include <hip/hip_runtime.h>

typedef __attribute__((ext_vector_type(2))) float v2f;
typedef __attribute__((ext_vector_type(4))) float v4f;
typedef __attribute__((ext_vector_type(8))) float v8f;

// ---------------------------------------------------------------------------
// Exclusive prefix sum of (a[i]+1) -> off[i].  Single block, 256 threads,
// blocked serial scan + Hillis-Steele over per-thread partials in LDS.
// n <= ~135K ints: trivial cost.
// ---------------------------------------------------------------------------
__global__ __launch_bounds__(256) void scan_offsets_kernel(
    const int* __restrict__ a, int n, int* __restrict__ off) {
  __shared__ int partial[256];
  const int t = threadIdx.x;
  const int chunk = (n + 255) / 256;
  const int begin = t * chunk;
  const int end = min(n, begin + chunk);
  int s = 0;
  for (int i = begin; i < end; ++i) s += a[i] + 1;
  partial[t] = s;
  __syncthreads();
  // Hillis-Steele inclusive scan over 256 partials
  for (int d = 1; d < 256; d <<= 1) {
    int v = (t >= d) ? partial[t - d] : 0;
    __syncthreads();
    partial[t] += v;
    __syncthreads();
  }
  int excl = (t == 0) ? 0 : partial[t - 1];
  for (int i = begin; i < end; ++i) {
    off[i] = excl;
    excl += a[i] + 1;
  }
}

// ---------------------------------------------------------------------------
// SAGE aggregation: out[r] = relu((src[r] + mean(neigh[off[r] .. +cnt])) * 0.5)
// One wave32 per row, D=128: each lane owns 4 contiguous floats (b128 loads).
// Neighbor stream is non-temporal (1.14 GB stream, don't pollute L2).
// ---------------------------------------------------------------------------
__global__ __launch_bounds__(256) void sage_kernel(
    const float* __restrict__ src, const float* __restrict__ neigh,
    const int* __restrict__ act, const int* __restrict__ off,
    float* __restrict__ out, int nRows) {
  const int wave = (int)((blockIdx.x * blockDim.x + threadIdx.x) >> 5);
  const int lane = threadIdx.x & 31;
  if (wave >= nRows) return;  // wave-uniform

  const int cnt = act[wave] + 1;
  const long segBase = (long)off[wave] * 32;  // in float4 units (128 floats/row)
  const v4f* __restrict__ np = (const v4f*)neigh + segBase + lane;

  v4f acc = (v4f)(0.0f);
  for (int j = 0; j < cnt; ++j) {
    v4f v = __builtin_nontemporal_load(np + (long)j * 32);
    acc += v;
  }
  const float inv = 1.0f / (float)cnt;
  const v4f s = ((const v4f*)src)[(long)wave * 32 + lane];
  v4f o = (s + acc * inv) * 0.5f;
  o.x = fmaxf(o.x, 0.0f);
  o.y = fmaxf(o.y, 0.0f);
  o.z = fmaxf(o.z, 0.0f);
  o.w = fmaxf(o.w, 0.0f);
  ((v4f*)out)[(long)wave * 32 + lane] = o;
}

// ---------------------------------------------------------------------------
// C = act(A[M,K] @ W[K,N] + bias), act = relu if RELU.
// One wave32 per 16x16 tile using V_WMMA_F32_16X16X4_F32, K stepped by 4.
// Fragment layouts per CDNA5 ISA 7.12.2:
//   A (16x4, 2 VGPRs): lanes 0-15 -> K={0,1}, lanes 16-31 -> K={2,3}, M=lane&15
//   B (4x16, 2 VGPRs): mirrored (K rows striped the same way, N=lane&15)
//   C/D (8 VGPRs):     VGPR j -> M=j (lanes 0-15) / M=j+8 (lanes 16-31), N=lane&15
// M,N multiples of 16 and exact grids => EXEC all-ones at every WMMA.
// ---------------------------------------------------------------------------
template <int RELU>
__global__ __launch_bounds__(256) void wmma_gemm_kernel(
    const float* __restrict__ A, const float* __restrict__ W,
    const float* __restrict__ bias, float* __restrict__ C,
    int M, int N, int K) {
  const int waveId = (int)((blockIdx.x * blockDim.x + threadIdx.x) >> 5);
  const int lane = threadIdx.x & 31;
  const int tilesN = N >> 4;
  const int tiles = (M >> 4) * tilesN;
  if (waveId >= tiles) return;  // wave-uniform

  const int tm = (waveId / tilesN) << 4;
  const int tn = (waveId % tilesN) << 4;
  const int half = lane >> 4;   // 0: lanes 0-15, 1: lanes 16-31
  const int idx = lane & 15;    // M index for A, N index for B/C/D

  v8f acc = {};
  const float* __restrict__ arow = A + (long)(tm + idx) * K;
  for (int k = 0; k < K; k += 4) {
    const int kk = k + 2 * half;
    v2f a, b;
    a.x = arow[kk];
    a.y = arow[kk + 1];
    b.x = W[(long)kk * N + tn + idx];
    b.y = W[(long)(kk + 1) * N + tn + idx];
    acc = __builtin_amdgcn_wmma_f32_16x16x4_f32(
        /*neg_a=*/false, a, /*neg_b=*/false, b,
        /*c_mod=*/(short)0, acc, /*reuse_a=*/false, /*reuse_b=*/false);
  }

  const float bn = bias[tn + idx];
#pragma unroll
  for (int j = 0; j < 8; ++j) {
    float v = acc[j] + bn;
    if (RELU) v = fmaxf(v, 0.0f);
    C[(long)(tm + j + 8 * half) * N + tn + idx] = v;
  }
}

// ---------------------------------------------------------------------------
// Host-side launcher
// ---------------------------------------------------------------------------
extern "C" void kernel_launch(void* const* d_in, const int* in_sizes, int n_in,
                              void* d_out, int out_size, void* d_ws, size_t ws_size,
                              hipStream_t stream) {
  const int*   actions0 = (const int*)d_in[0];
  const int*   actions1 = (const int*)d_in[1];
  const float* feat0    = (const float*)d_in[2];
  const float* feat1    = (const float*)d_in[3];
  const float* feat2    = (const float*)d_in[4];
  const float* W1       = (const float*)d_in[5];
  const float* b1       = (const float*)d_in[6];
  const float* W2       = (const float*)d_in[7];
  const float* b2       = (const float*)d_in[8];

  const int N0 = in_sizes[0];          // 8192
  const int N1 = in_sizes[1];          // 135168
  const int D_IN = 128, D_HID = 256, D_OUT = 64;

  // Workspace layout (256B aligned slices)
  char* ws = (char*)d_ws;
  auto align256 = [](size_t x) { return (x + 255) & ~(size_t)255; };
  size_t o = 0;
  int* off0 = (int*)(ws + o);   o = align256(o + (size_t)N0 * 4);
  int* off1 = (int*)(ws + o);   o = align256(o + (size_t)N1 * 4);
  float* h1  = (float*)(ws + o); o = align256(o + (size_t)N1 * D_IN * 4);
  float* h0a = (float*)(ws + o); o = align256(o + (size_t)N0 * D_IN * 4);
  float* H   = (float*)(ws + o); o = align256(o + (size_t)N0 * D_HID * 4);
  (void)ws_size; (void)n_in; (void)out_size;

  float* out_mlp = (float*)d_out;                       // [N0, 64]
  float* h0_out  = (float*)d_out + (size_t)N0 * D_OUT;  // [N0, 128]

  // 1) Segment offsets
  scan_offsets_kernel<<<1, 256, 0, stream>>>(actions0, N0, off0);
  scan_offsets_kernel<<<1, 256, 0, stream>>>(actions1, N1, off1);

  // 2) Layer 0
  {
    int blocks0 = (N0 + 7) / 8;   // 8 waves per 256-thread block
    int blocks1 = (N1 + 7) / 8;
    sage_kernel<<<blocks0, 256, 0, stream>>>(feat0, feat1, actions0, off0, h0a, N0);
    sage_kernel<<<blocks1, 256, 0, stream>>>(feat1, feat2, actions1, off1, h1, N1);
  }

  // 3) Layer 1 -> h0 written straight into d_out (also feeds the MLP)
  {
    int blocks0 = (N0 + 7) / 8;
    sage_kernel<<<blocks0, 256, 0, stream>>>(h0a, h1, actions0, off0, h0_out, N0);
  }

  // 4) MLP head with WMMA f32
  {
    int tiles1 = (N0 / 16) * (D_HID / 16);   // 8192 tiles
    int blocks1 = (tiles1 + 7) / 8;
    wmma_gemm_kernel<1><<<blocks1, 256, 0, stream>>>(h0_out, W1, b1, H, N0, D_HID, D_IN);

    int tiles2 = (N0 / 16) * (D_OUT / 16);   // 2048 tiles
    int blocks2 = (tiles2 + 7) / 8;
    wmma_gemm_kernel<0><<<blocks2, 256, 0, stream>>>(H, W2, b2, out_mlp, N0, D_OUT, D_HID);
  }
}